// tripletLoss_30554397344482
// MI455X (gfx1250) — compile-verified
//
#include <hip/hip_runtime.h>

typedef __attribute__((ext_vector_type(2))) float v2f;
typedef __attribute__((ext_vector_type(4))) float v4f;
typedef __attribute__((ext_vector_type(8))) float v8f;

#define NROW 8192
#define DDIM 512
#define EPSV 1e-8f
#define BT   128           // block tile (128x128 of C per workgroup)
#define NB   (NROW / BT)   // 64 block tiles per dimension
#define NBLK (NB * (NB + 1) / 2)  // 2080 upper-tri blocks

// ---------------------------------------------------------------------------
// Kernel 1: row norms.  One wave (32 lanes) per row, float4 loads, shuffle
// reduction.  8 rows per 256-thread block.
// ---------------------------------------------------------------------------
__global__ __launch_bounds__(256) void norms_kernel(const float* __restrict__ x,
                                                    float* __restrict__ norms) {
    const int wave = threadIdx.x >> 5;
    const int lane = threadIdx.x & 31;
    const int row  = blockIdx.x * 8 + wave;
    const float* xr = x + (size_t)row * DDIM;

    float s = 0.0f;
    #pragma unroll
    for (int p = 0; p < DDIM; p += 128) {
        v4f v = *(const v4f*)(xr + p + lane * 4);
        s += v.x * v.x + v.y * v.y + v.z * v.z + v.w * v.w;
    }
    #pragma unroll
    for (int off = 16; off > 0; off >>= 1)
        s += __shfl_xor(s, off, 32);
    if (lane == 0)
        norms[row] = sqrtf(s);
}

// ---------------------------------------------------------------------------
// Kernel 2: upper-triangular blocked X·Xt with V_WMMA_F32_16X16X4_F32.
// 128 threads = 4 waves; each wave owns a 64x64 C patch (4x4 tiles of 16x16).
// A and B fragments share the same per-lane addressing pattern on X.
// ---------------------------------------------------------------------------
__global__ __launch_bounds__(128) void sim_loss_kernel(const float* __restrict__ x,
                                                       const float* __restrict__ norms,
                                                       float* __restrict__ bsum,
                                                       unsigned* __restrict__ bcnt) {
    // Map linear block id -> (bi, bj) with bj >= bi (upper triangle of tiles).
    const int b = (int)blockIdx.x;
    int bi = (int)((2.0f * NB + 1.0f -
                    sqrtf((2.0f * NB + 1.0f) * (2.0f * NB + 1.0f) - 8.0f * (float)b)) * 0.5f);
    if (bi < 0) bi = 0;
    if (bi > NB - 1) bi = NB - 1;
    while ((bi + 1) * NB - ((bi + 1) * bi) / 2 <= b) ++bi;
    while (bi * NB - (bi * (bi - 1)) / 2 > b) --bi;
    const int bj = bi + (b - (bi * NB - (bi * (bi - 1)) / 2));

    const int wave = threadIdx.x >> 5;
    const int lane = threadIdx.x & 31;
    const int wm = wave >> 1;           // 2x2 wave grid inside the 128x128 block
    const int wn = wave & 1;
    const int row0 = bi * BT + wm * 64; // wave's 64-row strip of A side
    const int col0 = bj * BT + wn * 64; // wave's 64-row strip of B side (cols of C)

    // Fragment addressing (identical for A and B sides):
    //   lanes 0-15 : M/N = lane,    K = k+0,k+1
    //   lanes 16-31: M/N = lane-16, K = k+2,k+3
    const int lr = lane & 15;
    const int lk = (lane >> 4) << 1;
    const float* aptr = x + (size_t)(row0 + lr) * DDIM + lk;
    const float* bptr = x + (size_t)(col0 + lr) * DDIM + lk;

    v8f c[4][4];
    #pragma unroll
    for (int m = 0; m < 4; ++m)
        #pragma unroll
        for (int n = 0; n < 4; ++n)
            c[m][n] = (v8f){0.f, 0.f, 0.f, 0.f, 0.f, 0.f, 0.f, 0.f};

    for (int k = 0; k < DDIM; k += 4) {
        v2f af[4], bf[4];
        #pragma unroll
        for (int m = 0; m < 4; ++m)
            af[m] = *(const v2f*)(aptr + (size_t)(m * 16) * DDIM + k);
        #pragma unroll
        for (int n = 0; n < 4; ++n)
            bf[n] = *(const v2f*)(bptr + (size_t)(n * 16) * DDIM + k);
        #pragma unroll
        for (int m = 0; m < 4; ++m)
            #pragma unroll
            for (int n = 0; n < 4; ++n)
                c[m][n] = __builtin_amdgcn_wmma_f32_16x16x4_f32(
                    /*neg_a=*/false, af[m], /*neg_b=*/false, bf[n],
                    /*c_mod=*/(short)0, c[m][n],
                    /*reuse_a=*/false, /*reuse_b=*/false);
    }

    // Epilogue: C layout is VGPR v -> M = v + 8*(lane>>4), N = lane&15.
    const int jn    = lane & 15;
    const int ihalf = (lane >> 4) * 8;
    float    lsum = 0.0f;
    unsigned lcnt = 0;

    #pragma unroll
    for (int m = 0; m < 4; ++m) {
        const int gibase = row0 + m * 16 + ihalf;
        float niv[8];
        #pragma unroll
        for (int v = 0; v < 8; ++v)
            niv[v] = norms[gibase + v];
        #pragma unroll
        for (int n = 0; n < 4; ++n) {
            const int   gj = col0 + n * 16 + jn;
            const float nj = norms[gj];
            #pragma unroll
            for (int v = 0; v < 8; ++v) {
                const int gi = gibase + v;
                if (gj >= gi) {
                    const float sim = c[m][n][v] / fmaxf(niv[v] * nj, EPSV);
                    const float L = ((gi >> 2) == (gj >> 2))
                                        ? (1.0f - sim)
                                        : fmaxf(sim - 1.0f, 0.0f);
                    lsum += L;
                    if (L != 0.0f) ++lcnt;
                }
            }
        }
    }

    // Wave reduce, then block reduce via LDS; one partial per block.
    #pragma unroll
    for (int off = 16; off > 0; off >>= 1) {
        lsum += __shfl_xor(lsum, off, 32);
        lcnt += (unsigned)__shfl_xor((int)lcnt, off, 32);
    }
    __shared__ float    swsum[4];
    __shared__ unsigned swcnt[4];
    if (lane == 0) { swsum[wave] = lsum; swcnt[wave] = lcnt; }
    __syncthreads();
    if (threadIdx.x == 0) {
        bsum[blockIdx.x] = swsum[0] + swsum[1] + swsum[2] + swsum[3];
        bcnt[blockIdx.x] = swcnt[0] + swcnt[1] + swcnt[2] + swcnt[3];
    }
}

// ---------------------------------------------------------------------------
// Kernel 3: deterministic final reduction of the 2080 block partials.
// ---------------------------------------------------------------------------
__global__ __launch_bounds__(256) void finalize_kernel(const float* __restrict__ bsum,
                                                       const unsigned* __restrict__ bcnt,
                                                       int nblocks,
                                                       float* __restrict__ out) {
    __shared__ float    ss[256];
    __shared__ unsigned sc[256];
    float    s = 0.0f;
    unsigned c = 0;
    for (int i = threadIdx.x; i < nblocks; i += 256) {
        s += bsum[i];
        c += bcnt[i];
    }
    ss[threadIdx.x] = s;
    sc[threadIdx.x] = c;
    __syncthreads();
    for (int off = 128; off > 0; off >>= 1) {
        if ((int)threadIdx.x < off) {
            ss[threadIdx.x] += ss[threadIdx.x + off];
            sc[threadIdx.x] += sc[threadIdx.x + off];
        }
        __syncthreads();
    }
    if (threadIdx.x == 0) {
        const float loss = sc[0] ? (ss[0] / (float)sc[0]) : 0.0f;
        out[0] = (loss == 0.0f) ? 0.0f : loss;
    }
}

// ---------------------------------------------------------------------------
extern "C" void kernel_launch(void* const* d_in, const int* in_sizes, int n_in,
                              void* d_out, int out_size, void* d_ws, size_t ws_size,
                              hipStream_t stream) {
    (void)in_sizes; (void)n_in; (void)out_size; (void)ws_size;
    const float* x = (const float*)d_in[0];

    float*    norms = (float*)d_ws;           // [8192]
    float*    bsum  = norms + NROW;           // [2080]
    unsigned* bcnt  = (unsigned*)(bsum + NBLK); // [2080]

    norms_kernel<<<NROW / 8, 256, 0, stream>>>(x, norms);
    sim_loss_kernel<<<NBLK, 128, 0, stream>>>(x, norms, bsum, bcnt);
    finalize_kernel<<<1, 256, 0, stream>>>(bsum, bcnt, NBLK, (float*)d_out);
}